// IPGR_5703716569302
// MI455X (gfx1250) — compile-verified
//
#include <hip/hip_runtime.h>
#include <hip/hip_bf16.h>
#include <stddef.h>

// Problem constants (match reference)
#define BATCH 8
#define NPTS  8192
#define MPTS  2048
#define EPSV  1e-6f
#define BASE_ALPHA 0.05f

typedef __attribute__((ext_vector_type(2))) float v2f;
typedef __attribute__((ext_vector_type(8))) float v8f;

// ---------------------------------------------------------------------------
// Phase A: per pred point, min squared distance + argmin over partial points.
//
// Trick: for fixed pred row i, argmin_j d2(i,j) == argmin_j (b2_j - 2 a_i.b_j),
// since a2_i is a constant shift. Encode that entire expression in one
// V_WMMA_F32_16X16X4_F32:
//    A row i  = (-2x_i, -2y_i, -2z_i, 1)      (16x4)
//    B col j  = ( x_j,   y_j,   z_j,  b2_j)   (4x16, b2 precomputed in LDS)
//    D[i][j]  = b2_j - 2 a_i.b_j              (the comparison key)
// The hot loop is then just: LDS b128 load -> 2 cndmask -> WMMA -> 8 cmp+sel.
//
// VGPR layouts (ISA 7.12.2, f32 16x16x4):
//   A: lane L holds row M=L&15; lanes 0-15 K={0,1}, lanes 16-31 K={2,3}.
//   B: lane L holds col N=L&15; same K split.
//   C/D: lane L holds col N=L&15; VGPR r holds row M=(L>=16?8:0)+r.
// ---------------------------------------------------------------------------
__global__ __launch_bounds__(256) void knn_min_wmma(
    const float* __restrict__ refined,         // [B,N,3]
    const float* __restrict__ partial,         // [B,M,3]
    float*       __restrict__ min_d2,          // [B*N]
    int*         __restrict__ min_idx,         // [B*N]
    unsigned int* __restrict__ batch_max_bits) // [B], pre-zeroed
{
  __shared__ float4 sp[MPTS];                  // 32 KB: (x, y, z, b2)

  const int tid  = threadIdx.x;
  const int lane = tid & 31;
  const int wave = tid >> 5;
  const int blk  = blockIdx.x;                 // 512 blocks
  const int b    = blk >> 6;                   // 64 blocks per batch
  const int grp  = blk & 63;

  // Stage this batch's partial points into LDS with b2 precomputed.
  for (int p = tid; p < MPTS; p += 256) {
    const float* src = partial + ((size_t)b * MPTS + p) * 3;
    const float x = src[0], y = src[1], z = src[2];
    sp[p] = make_float4(x, y, z, x * x + y * y + z * z);
  }
  __syncthreads();

  const int n_base = (grp * 8 + wave) * 16;
  const int row    = lane & 15;                // pred row (for A) / D column
  const int hi     = lane >> 4;                // half-wave selector (K split)

  // A operand: (-2x,-2y) on low lanes, (-2z, 1) on high lanes.
  const float* ap = refined + ((size_t)b * NPTS + n_base + row) * 3;
  const float ax = ap[0], ay = ap[1], az = ap[2];
  const float a2 = ax * ax + ay * ay + az * az;
  v2f a;
  a.x = hi ? (-2.0f * az) : (-2.0f * ax);
  a.y = hi ? 1.0f         : (-2.0f * ay);

  // |a|^2 for the 8 rows this lane accumulates in D (rows hi*8+r): one-time.
  float a2r[8];
  const int sbase = hi << 3;
#pragma unroll
  for (int r = 0; r < 8; ++r) a2r[r] = __shfl(a2, sbase + r, 32);

  float mind[8];
  int   midx[8];
#pragma unroll
  for (int r = 0; r < 8; ++r) { mind[r] = 3.4e38f; midx[r] = 0; }

  // Sweep partial points: 2 tiles (2 WMMAs) per iteration, 64 iterations.
  for (int m0 = 0; m0 < MPTS; m0 += 32) {
    const int j0 = m0 + row;
    const int j1 = j0 + 16;
    const float4 p0 = sp[j0];
    const float4 p1 = sp[j1];

    v2f b0, b1;
    b0.x = hi ? p0.z : p0.x;  b0.y = hi ? p0.w : p0.y;
    b1.x = hi ? p1.z : p1.x;  b1.y = hi ? p1.w : p1.y;

    v8f c = {};
    const v8f d0 = __builtin_amdgcn_wmma_f32_16x16x4_f32(
        false, a, false, b0, (short)0, c, false, false);
    const v8f d1 = __builtin_amdgcn_wmma_f32_16x16x4_f32(
        false, a, false, b1, (short)0, c, false, false);

#pragma unroll
    for (int r = 0; r < 8; ++r) {
      const bool t0 = d0[r] < mind[r];
      mind[r] = t0 ? d0[r] : mind[r];
      midx[r] = t0 ? j0    : midx[r];
    }
#pragma unroll
    for (int r = 0; r < 8; ++r) {
      const bool t1 = d1[r] < mind[r];
      mind[r] = t1 ? d1[r] : mind[r];
      midx[r] = t1 ? j1    : midx[r];
    }
  }

  // Reduce running min over the 16 columns held by each half-wave (branchless).
#pragma unroll
  for (int r = 0; r < 8; ++r) {
#pragma unroll
    for (int mask = 1; mask <= 8; mask <<= 1) {
      const float od = __shfl_xor(mind[r], mask, 32);
      const int   oi = __shfl_xor(midx[r], mask, 32);
      const bool take = od < mind[r];
      mind[r] = take ? od : mind[r];
      midx[r] = take ? oi : midx[r];
    }
  }

  // Lanes 0 and 16 hold reduced results for rows 0-7 / 8-15 of the tile.
  if (row == 0) {
    float wmax = 0.0f;
#pragma unroll
    for (int r = 0; r < 8; ++r) {
      const int i = n_base + (hi << 3) + r;
      const float d2c = fmaxf(mind[r] + a2r[r], 0.0f);  // restore a2, clamp
      min_d2[(size_t)b * NPTS + i]  = d2c;
      min_idx[(size_t)b * NPTS + i] = midx[r];
      wmax = fmaxf(wmax, d2c);
    }
    // d2 >= 0 so float bit pattern ordering == numeric ordering.
    atomicMax(&batch_max_bits[b], __float_as_uint(wmax));
  }
}

// ---------------------------------------------------------------------------
// Phase B: refined = refined + alpha * (nearest - refined)
// alpha = 0.05 * (2 - sqrt(min_d2)/(sqrt(max_d2_batch) + eps))
// ---------------------------------------------------------------------------
__global__ __launch_bounds__(256) void refine_update(
    const float* __restrict__ refined_in,      // [B,N,3]
    const float* __restrict__ partial,         // [B,M,3]
    const float* __restrict__ min_d2,          // [B*N]
    const int*   __restrict__ min_idx,         // [B*N]
    const unsigned int* __restrict__ batch_max_bits, // [B]
    float*       __restrict__ refined_out)     // [B,N,3]
{
  const int gid = blockIdx.x * 256 + threadIdx.x;
  if (gid >= BATCH * NPTS) return;
  const int b = gid / NPTS;

  const float mx = sqrtf(__uint_as_float(batch_max_bits[b]));
  const float md = sqrtf(fmaxf(min_d2[gid], 0.0f));
  const float dist_norm = md / (mx + EPSV);
  const float alpha = BASE_ALPHA * (2.0f - dist_norm);

  const int j = min_idx[gid];
  const float* np = partial + ((size_t)b * MPTS + j) * 3;
  const float* rp = refined_in + (size_t)gid * 3;
  float*       op = refined_out + (size_t)gid * 3;

  const float r0 = rp[0], r1 = rp[1], r2 = rp[2];
  op[0] = r0 + alpha * (np[0] - r0);
  op[1] = r1 + alpha * (np[1] - r1);
  op[2] = r2 + alpha * (np[2] - r2);
}

extern "C" void kernel_launch(void* const* d_in, const int* in_sizes, int n_in,
                              void* d_out, int out_size, void* d_ws, size_t ws_size,
                              hipStream_t stream) {
  (void)in_sizes; (void)n_in; (void)out_size; (void)ws_size;

  const float* pred    = (const float*)d_in[0];   // [8,8192,3] f32
  const float* partial = (const float*)d_in[1];   // [8,2048,3] f32
  float* out = (float*)d_out;                     // [8,8192,3] f32

  // Workspace layout: min_d2 (256KB) | min_idx (256KB) | batch_max (32B)
  float* ws_min_d2 = (float*)d_ws;
  int*   ws_idx    = (int*)((char*)d_ws + (size_t)BATCH * NPTS * sizeof(float));
  unsigned int* ws_max =
      (unsigned int*)((char*)d_ws + 2u * (size_t)BATCH * NPTS * sizeof(float));

  const dim3 gA(512), bA(256);                               // 8 waves/block
  const dim3 gB((BATCH * NPTS + 255) / 256), bB(256);

  // Iteration 1: refined = pred
  hipMemsetAsync(ws_max, 0, BATCH * sizeof(unsigned int), stream);
  knn_min_wmma<<<gA, bA, 0, stream>>>(pred, partial, ws_min_d2, ws_idx, ws_max);
  refine_update<<<gB, bB, 0, stream>>>(pred, partial, ws_min_d2, ws_idx, ws_max, out);

  // Iteration 2: refined = out (in-place update, per-thread elementwise)
  hipMemsetAsync(ws_max, 0, BATCH * sizeof(unsigned int), stream);
  knn_min_wmma<<<gA, bA, 0, stream>>>(out, partial, ws_min_d2, ws_idx, ws_max);
  refine_update<<<gB, bB, 0, stream>>>(out, partial, ws_min_d2, ws_idx, ws_max, out);
}